// DLI_loss_2_6614249636366
// MI455X (gfx1250) — compile-verified
//
#include <hip/hip_runtime.h>
#include <math.h>

typedef __attribute__((ext_vector_type(2))) float v2f;
typedef __attribute__((ext_vector_type(8))) float v8f;

#define B_  64
#define T_  128
#define D_  1024
#define H_  512
#define J_  126          // T-2
#define M_  (B_ * J_)    // 8064 rows; 8064 = 252*32

#define WMMA_F32(a, b, c) \
    __builtin_amdgcn_wmma_f32_16x16x4_f32(false, (a), false, (b), (short)0, (c), false, false)

__device__ __forceinline__ float sig_(float x) { return 1.0f / (1.0f + expf(-x)); }

// ---------------------------------------------------------------- lengths[b]
__global__ void k_lengths(const int* __restrict__ mask, int* __restrict__ lengths) {
    int b = threadIdx.x;
    if (b < B_) {
        int s = 0;
        for (int t = 0; t < T_; ++t) s += mask[b * T_ + t];
        lengths[b] = s;
    }
}

// ------------------------------------------------- xs[b,t] = x[b,t,:] . w_x
__global__ __launch_bounds__(256) void k_xs(const float* __restrict__ x,
                                            const float* __restrict__ fc_w,
                                            float* __restrict__ xs) {
    int wave = threadIdx.x >> 5, lane = threadIdx.x & 31;
    int row = blockIdx.x * 8 + wave;              // 0 .. B*T-1
    const float* xr = x + (size_t)row * D_;
    const float* wx = fc_w + H_;
    float s = 0.f;
    for (int k = lane; k < D_; k += 32) s += xr[k] * wx[k];
    for (int o = 16; o >= 1; o >>= 1) s += __shfl_xor(s, o, 32);
    if (lane == 0) xs[row] = s;
}

// --------------------------------------------- LSTM step 1 (h0 = c0 = 0)
// Block = 4 waves (one gate each). Tile: 32 rows x 64 h-cols.
// Per wave: 8 independent f32-WMMA accumulators (2 m-subtiles x 4 n-subtiles).
__global__ __launch_bounds__(128) void k_lstm1(const float* __restrict__ x,
                                               const float* __restrict__ W_ih,
                                               const float* __restrict__ b_ih,
                                               const float* __restrict__ b_hh,
                                               float* __restrict__ h1,
                                               float* __restrict__ c1) {
    __shared__ float tiles[4][32][65];
    const int tid  = threadIdx.x;
    const int g    = tid >> 5, lane = tid & 31;
    const int m0   = blockIdx.x * 32, n0 = blockIdx.y * 64;
    const int ml   = lane & 15;
    const int kh   = (lane >> 4) * 2;             // lanes 16-31 hold K=2,3
    const int mA   = m0 + ml, mB = m0 + 16 + ml;
    const int bA   = mA / J_, jA = mA % J_;
    const int bB   = mB / J_, jB = mB % J_;
    const float* arowA = x + ((size_t)bA * T_ + jA) * D_ + kh;
    const float* arowB = x + ((size_t)bB * T_ + jB) * D_ + kh;
    const float* brow  = W_ih + (size_t)(g * H_ + n0 + ml) * D_ + kh;

    v8f acc[2][4] = {};
#pragma unroll 2
    for (int k0 = 0; k0 < D_; k0 += 4) {
        v2f aA = *(const v2f*)(arowA + k0);
        v2f aB = *(const v2f*)(arowB + k0);
        v2f b0 = *(const v2f*)(brow + k0);
        v2f b1 = *(const v2f*)(brow + k0 + (size_t)16 * D_);
        v2f b2 = *(const v2f*)(brow + k0 + (size_t)32 * D_);
        v2f b3 = *(const v2f*)(brow + k0 + (size_t)48 * D_);
        acc[0][0] = WMMA_F32(aA, b0, acc[0][0]);
        acc[0][1] = WMMA_F32(aA, b1, acc[0][1]);
        acc[0][2] = WMMA_F32(aA, b2, acc[0][2]);
        acc[0][3] = WMMA_F32(aA, b3, acc[0][3]);
        acc[1][0] = WMMA_F32(aB, b0, acc[1][0]);
        acc[1][1] = WMMA_F32(aB, b1, acc[1][1]);
        acc[1][2] = WMMA_F32(aB, b2, acc[1][2]);
        acc[1][3] = WMMA_F32(aB, b3, acc[1][3]);
    }
    const int mro = (lane >> 4) * 8;              // C/D layout: lanes 16-31 -> M+8
#pragma unroll
    for (int s = 0; s < 4; ++s) {
        int col  = g * H_ + n0 + s * 16 + ml;
        float bv = b_ih[col] + b_hh[col];
#pragma unroll
        for (int r = 0; r < 8; ++r) {
            tiles[g][mro + r][s * 16 + ml]      = acc[0][s][r] + bv;
            tiles[g][16 + mro + r][s * 16 + ml] = acc[1][s][r] + bv;
        }
    }
    __syncthreads();

    for (int e = tid; e < 32 * 64; e += 128) {
        int mm = e >> 6, nn = e & 63;
        float iv = tiles[0][mm][nn];
        float gv = tiles[2][mm][nn];
        float ov = tiles[3][mm][nn];
        float cc = sig_(iv) * tanhf(gv);          // sig(f)*c0 term is zero
        float hh = sig_(ov) * tanhf(cc);
        size_t idx = (size_t)(m0 + mm) * H_ + (n0 + nn);
        h1[idx] = hh;
        c1[idx] = cc;
    }
}

// --------------------------------------------- LSTM step 2 + h2 . w_h partials
__global__ __launch_bounds__(128) void k_lstm2(const float* __restrict__ x,
                                               const float* __restrict__ W_ih,
                                               const float* __restrict__ W_hh,
                                               const float* __restrict__ b_ih,
                                               const float* __restrict__ b_hh,
                                               const float* __restrict__ h1,
                                               const float* __restrict__ c1,
                                               const float* __restrict__ fc_w,
                                               float* __restrict__ hpart) {
    __shared__ float tiles[4][32][65];
    const int tid  = threadIdx.x;
    const int g    = tid >> 5, lane = tid & 31;
    const int m0   = blockIdx.x * 32, n0 = blockIdx.y * 64;
    const int ml   = lane & 15;
    const int kh   = (lane >> 4) * 2;
    const int mA   = m0 + ml, mB = m0 + 16 + ml;
    const int bA   = mA / J_, jA = mA % J_;
    const int bB   = mB / J_, jB = mB % J_;
    const float* arowA = x + ((size_t)bA * T_ + (jA + 1)) * D_ + kh;   // X1
    const float* arowB = x + ((size_t)bB * T_ + (jB + 1)) * D_ + kh;
    const float* brow  = W_ih + (size_t)(g * H_ + n0 + ml) * D_ + kh;
    const float* hrowA = h1 + (size_t)mA * H_ + kh;
    const float* hrowB = h1 + (size_t)mB * H_ + kh;
    const float* brow2 = W_hh + (size_t)(g * H_ + n0 + ml) * H_ + kh;

    v8f acc[2][4] = {};
#pragma unroll 2
    for (int k0 = 0; k0 < D_; k0 += 4) {
        v2f aA = *(const v2f*)(arowA + k0);
        v2f aB = *(const v2f*)(arowB + k0);
        v2f b0 = *(const v2f*)(brow + k0);
        v2f b1 = *(const v2f*)(brow + k0 + (size_t)16 * D_);
        v2f b2 = *(const v2f*)(brow + k0 + (size_t)32 * D_);
        v2f b3 = *(const v2f*)(brow + k0 + (size_t)48 * D_);
        acc[0][0] = WMMA_F32(aA, b0, acc[0][0]);
        acc[0][1] = WMMA_F32(aA, b1, acc[0][1]);
        acc[0][2] = WMMA_F32(aA, b2, acc[0][2]);
        acc[0][3] = WMMA_F32(aA, b3, acc[0][3]);
        acc[1][0] = WMMA_F32(aB, b0, acc[1][0]);
        acc[1][1] = WMMA_F32(aB, b1, acc[1][1]);
        acc[1][2] = WMMA_F32(aB, b2, acc[1][2]);
        acc[1][3] = WMMA_F32(aB, b3, acc[1][3]);
    }
#pragma unroll 2
    for (int k0 = 0; k0 < H_; k0 += 4) {
        v2f aA = *(const v2f*)(hrowA + k0);
        v2f aB = *(const v2f*)(hrowB + k0);
        v2f b0 = *(const v2f*)(brow2 + k0);
        v2f b1 = *(const v2f*)(brow2 + k0 + (size_t)16 * H_);
        v2f b2 = *(const v2f*)(brow2 + k0 + (size_t)32 * H_);
        v2f b3 = *(const v2f*)(brow2 + k0 + (size_t)48 * H_);
        acc[0][0] = WMMA_F32(aA, b0, acc[0][0]);
        acc[0][1] = WMMA_F32(aA, b1, acc[0][1]);
        acc[0][2] = WMMA_F32(aA, b2, acc[0][2]);
        acc[0][3] = WMMA_F32(aA, b3, acc[0][3]);
        acc[1][0] = WMMA_F32(aB, b0, acc[1][0]);
        acc[1][1] = WMMA_F32(aB, b1, acc[1][1]);
        acc[1][2] = WMMA_F32(aB, b2, acc[1][2]);
        acc[1][3] = WMMA_F32(aB, b3, acc[1][3]);
    }
    const int mro = (lane >> 4) * 8;
#pragma unroll
    for (int s = 0; s < 4; ++s) {
        int col  = g * H_ + n0 + s * 16 + ml;
        float bv = b_ih[col] + b_hh[col];
#pragma unroll
        for (int r = 0; r < 8; ++r) {
            tiles[g][mro + r][s * 16 + ml]      = acc[0][s][r] + bv;
            tiles[g][16 + mro + r][s * 16 + ml] = acc[1][s][r] + bv;
        }
    }
    __syncthreads();

    for (int e = tid; e < 32 * 64; e += 128) {
        int mm = e >> 6, nn = e & 63;
        float iv = tiles[0][mm][nn];
        float fv = tiles[1][mm][nn];
        float gv = tiles[2][mm][nn];
        float ov = tiles[3][mm][nn];
        float c0 = c1[(size_t)(m0 + mm) * H_ + (n0 + nn)];
        float cc = sig_(fv) * c0 + sig_(iv) * tanhf(gv);
        float h2 = sig_(ov) * tanhf(cc);
        tiles[0][mm][nn] = h2 * fc_w[n0 + nn];    // own slot only: no race
    }
    __syncthreads();
    if (tid < 32) {                               // per-row partial of hs
        float s = 0.f;
#pragma unroll
        for (int nn = 0; nn < 64; ++nn) s += tiles[0][tid][nn];
        hpart[(size_t)(m0 + tid) * 8 + blockIdx.y] = s;
    }
}

// --------------------------------------------- masked logsumexp loss per row
__global__ __launch_bounds__(256) void k_loss(const float* __restrict__ xs,
                                              const float* __restrict__ hpart,
                                              const int* __restrict__ lengths,
                                              const float* __restrict__ fc_b_p,
                                              float* __restrict__ loss_row,
                                              float* __restrict__ validf) {
    int wave = threadIdx.x >> 5, lane = threadIdx.x & 31;
    int row = blockIdx.x * 8 + wave;              // 0 .. M_-1
    if (row >= M_) return;
    int bb = row / J_, j = row % J_;
    int len = lengths[bb];

    float hs = (lane < 8) ? hpart[(size_t)row * 8 + lane] : 0.f;
    for (int o = 16; o >= 1; o >>= 1) hs += __shfl_xor(hs, o, 32);

    const float* xrow = xs + bb * T_;
    const float base = hs + fc_b_p[0];
    float vals[4];
    float lmax = -1.0e9f;
#pragma unroll
    for (int q = 0; q < 4; ++q) {
        int r = lane + q * 32;
        float v = -1.0e9f;
        if (r < J_) {
            int pos = j + 2 + r;
            int pc = pos < T_ ? pos : (T_ - 1);
            v = base + xrow[pc];
            if (pos >= len) v = -1.0e9f;          // reference mask value
        }
        vals[q] = v;
        lmax = fmaxf(lmax, v);
    }
    for (int o = 16; o >= 1; o >>= 1) lmax = fmaxf(lmax, __shfl_xor(lmax, o, 32));
    float se = 0.f;
#pragma unroll
    for (int q = 0; q < 4; ++q) se += expf(vals[q] - lmax);
    for (int o = 16; o >= 1; o >>= 1) se += __shfl_xor(se, o, 32);
    float lse = lmax + logf(se);
    float logit0 = base + xrow[j + 2];            // r=0; unmasked iff row valid
    int valid = (j + 3) <= len;
    if (lane == 0) {
        loss_row[row] = valid ? (lse - logit0) : 0.f;
        validf[row]   = valid ? 1.f : 0.f;
    }
}

// --------------------------------------------- final mean over valid rows
__global__ __launch_bounds__(256) void k_final(const float* __restrict__ loss_row,
                                               const float* __restrict__ validf,
                                               float* __restrict__ out) {
    __shared__ float sl[256], sc[256];
    int tid = threadIdx.x;
    float s = 0.f, c = 0.f;
    for (int i = tid; i < M_; i += 256) { s += loss_row[i]; c += validf[i]; }
    sl[tid] = s; sc[tid] = c;
    __syncthreads();
    for (int o = 128; o >= 1; o >>= 1) {
        if (tid < o) { sl[tid] += sl[tid + o]; sc[tid] += sc[tid + o]; }
        __syncthreads();
    }
    if (tid == 0) {
        float cnt = sc[0];
        out[0] = (cnt > 0.f) ? (sl[0] / cnt) : 0.f;
    }
}

extern "C" void kernel_launch(void* const* d_in, const int* in_sizes, int n_in,
                              void* d_out, int out_size, void* d_ws, size_t ws_size,
                              hipStream_t stream) {
    (void)in_sizes; (void)n_in; (void)out_size; (void)ws_size;
    const float* x    = (const float*)d_in[0];   // (B,T,D)
    const int*   mask = (const int*)  d_in[1];   // (B,T)
    const float* W_ih = (const float*)d_in[2];   // (4H,D)
    const float* W_hh = (const float*)d_in[3];   // (4H,H)
    const float* b_ih = (const float*)d_in[4];   // (4H,)
    const float* b_hh = (const float*)d_in[5];   // (4H,)
    const float* fc_w = (const float*)d_in[6];   // (H+D,)
    const float* fc_b = (const float*)d_in[7];   // scalar

    float* ws      = (float*)d_ws;
    int*   lengths = (int*)ws;                   // 64 ints
    float* xs      = ws + 64;                    // 8192
    float* hpart   = ws + 8256;                  // 8064*8 = 64512
    float* lrow    = ws + 72768;                 // 8064
    float* vrow    = ws + 80832;                 // 8064
    float* h1      = ws + 88896;                 // 8064*512
    float* c1      = ws + 4217664;               // 8064*512 (ends ~33.4 MB)

    k_lengths<<<1, 64, 0, stream>>>(mask, lengths);
    k_xs<<<(B_ * T_) / 8, 256, 0, stream>>>(x, fc_w, xs);
    k_lstm1<<<dim3(M_ / 32, H_ / 64), 128, 0, stream>>>(x, W_ih, b_ih, b_hh, h1, c1);
    k_lstm2<<<dim3(M_ / 32, H_ / 64), 128, 0, stream>>>(x, W_ih, W_hh, b_ih, b_hh,
                                                        h1, c1, fc_w, hpart);
    k_loss<<<M_ / 8, 256, 0, stream>>>(xs, hpart, lengths, fc_b, lrow, vrow);
    k_final<<<1, 256, 0, stream>>>(lrow, vrow, (float*)d_out);
}